// TDHConv_1932735283953
// MI455X (gfx1250) — compile-verified
//
#include <hip/hip_runtime.h>
#include <hip/hip_bf16.h>
#include <math.h>

// ---------------------------------------------------------------------------
// TDHConv for MI455X (gfx1250, wave32).
//  - nv / ee linear layers: V_WMMA_F32_16X16X4_F32 (exact f32 matmul on matrix cores)
//  - phase B (edge aggregation): wave-per-edge online softmax, zero atomics
//  - phase C (node propagation): wave-per-incidence + u32-encoded float atomicMax
//    and native f32 atomic adds
// ---------------------------------------------------------------------------

typedef __attribute__((ext_vector_type(2))) float v2f;
typedef __attribute__((ext_vector_type(8))) float v8f;

#define DDIM 256
#define TDIM 128

__device__ __forceinline__ float wred(float v) {
  v += __shfl_xor(v, 16, 32);
  v += __shfl_xor(v, 8, 32);
  v += __shfl_xor(v, 4, 32);
  v += __shfl_xor(v, 2, 32);
  v += __shfl_xor(v, 1, 32);
  return v;
}

// monotone uint encoding of float for atomicMax (init value 0 == -inf identity)
__device__ __forceinline__ unsigned fkey(float f) {
  unsigned u = __float_as_uint(f);
  return (u & 0x80000000u) ? ~u : (u | 0x80000000u);
}
__device__ __forceinline__ float fdec(unsigned k) {
  return __uint_as_float((k & 0x80000000u) ? (k ^ 0x80000000u) : ~k);
}

__device__ __forceinline__ int lbound(const int* __restrict__ a, int n, int key) {
  int lo = 0, hi = n;
  while (lo < hi) {
    int mid = (lo + hi) >> 1;
    if (a[mid] < key) lo = mid + 1; else hi = mid;
  }
  return lo;
}

// ---------------------------------------------------------------------------
// C[nRows x 256] = A[nRows x 256] * W[256 x 256]^T + bias
// One wave computes a 16x64 stripe of C via 4 accumulators of
// V_WMMA_F32_16X16X4_F32.  A-frag / B-frag: lane%16 -> M/N index,
// (lane/16)*2 + {0,1} -> K within the 4-wide step (float2 per lane).
// ---------------------------------------------------------------------------
__global__ __launch_bounds__(256)
void gemm256_wmma_kernel(const float* __restrict__ A, const float* __restrict__ W,
                         const float* __restrict__ bias, float* __restrict__ C,
                         int nRows) {
  const int lane = threadIdx.x & 31;
  const int gw = blockIdx.x * 8 + (threadIdx.x >> 5);
  const int rowTile = gw >> 2;   // 4 column-quads of 64 per row tile
  const int colQ = gw & 3;
  if (rowTile * 16 >= nRows) return;
  const int l16 = lane & 15;
  const int half = lane >> 4;

  const float* aPtr = A + (size_t)(rowTile * 16 + l16) * DDIM + half * 2;
  const float* wPtr = W + (size_t)(colQ * 64 + l16) * DDIM + half * 2;

  v8f acc[4] = {v8f{}, v8f{}, v8f{}, v8f{}};

#pragma unroll 4
  for (int k = 0; k < DDIM; k += 4) {
    v2f a = *(const v2f*)(aPtr + k);
    v2f b0 = *(const v2f*)(wPtr + k);
    v2f b1 = *(const v2f*)(wPtr + 16 * DDIM + k);
    v2f b2 = *(const v2f*)(wPtr + 32 * DDIM + k);
    v2f b3 = *(const v2f*)(wPtr + 48 * DDIM + k);
    acc[0] = __builtin_amdgcn_wmma_f32_16x16x4_f32(false, a, false, b0, (short)0, acc[0], false, false);
    acc[1] = __builtin_amdgcn_wmma_f32_16x16x4_f32(false, a, false, b1, (short)0, acc[1], false, false);
    acc[2] = __builtin_amdgcn_wmma_f32_16x16x4_f32(false, a, false, b2, (short)0, acc[2], false, false);
    acc[3] = __builtin_amdgcn_wmma_f32_16x16x4_f32(false, a, false, b3, (short)0, acc[3], false, false);
  }

  // C/D layout: lanes 0-15 -> N=l16, M = r (vgpr r); lanes 16-31 -> M = 8 + r
#pragma unroll
  for (int j = 0; j < 4; j++) {
    const int col = colQ * 64 + j * 16 + l16;
    const float bv = bias[col];
    float* cp = C + (size_t)(rowTile * 16 + half * 8) * DDIM + col;
#pragma unroll
    for (int r = 0; r < 8; r++) cp[(size_t)r * DDIM] = acc[j][r] + bv;
  }
}

// ---------------------------------------------------------------------------
// Phase B: one wave per edge.  Online softmax over the edge's (sorted)
// incidence segment; result written directly (no atomics):
//   in : enew[eg] = 2*edge_emb[eg] + sum(attn * node_emb)
//   out: enew[eg] = -sum(attn * node_emb)
// ---------------------------------------------------------------------------
__global__ __launch_bounds__(256)
void edge_aggr_kernel(const float* __restrict__ edge_emb, const float* __restrict__ ee,
                      const float* __restrict__ nv, const float* __restrict__ nemb,
                      const float* __restrict__ alpha_w,
                      const int* __restrict__ nidx, const int* __restrict__ eseg,
                      const int* __restrict__ emap, float* __restrict__ enew,
                      int nEdges, int M, int isOut) {
  const int e = blockIdx.x * 8 + (threadIdx.x >> 5);
  if (e >= nEdges) return;
  const int lane = threadIdx.x & 31;
  const int eg = emap[e];
  const int start = lbound(eseg, M, e);
  const int end = lbound(eseg, M, e + 1);

  float al[8], ev[8], acc[8];
#pragma unroll
  for (int j = 0; j < 8; j++) {
    const int d = lane + 32 * j;
    al[j] = alpha_w[d];
    ev[j] = ee[(size_t)eg * DDIM + d];
    acc[j] = 0.f;
  }
  float mx = -__builtin_inff();
  float s = 0.f;

  for (int m = start; m < end; m++) {
    const int n = nidx[m];
    const float* nvp = nv + (size_t)n * DDIM;
    float p = 0.f;
#pragma unroll
    for (int j = 0; j < 8; j++) {
      float x = ev[j] + nvp[lane + 32 * j];
      x = (x >= 0.f) ? x : 0.01f * x;          // LeakyReLU(0.01)
      p = fmaf(al[j], x, p);
    }
    p = wred(p);                               // all lanes hold the logit
    float f = 1.f;
    if (p > mx) { f = __expf(mx - p); mx = p; }  // first iter: exp(-inf)=0
    const float et = __expf(p - mx);
    s = s * f + et;
    const float* xp = nemb + (size_t)n * DDIM;
#pragma unroll
    for (int j = 0; j < 8; j++)
      acc[j] = fmaf(acc[j], f, et * xp[lane + 32 * j]);
  }

  const float inv = (s > 0.f) ? 1.f / s : 0.f;
  const float sign = isOut ? -1.f : 1.f;
#pragma unroll
  for (int j = 0; j < 8; j++) {
    const int d = lane + 32 * j;
    const float base = isOut ? 0.f : 2.f * edge_emb[(size_t)eg * DDIM + d];
    enew[(size_t)eg * DDIM + d] = base + sign * acc[j] * inv;
  }
}

// out[i, t] = cos(time[i] * w[t] + b[t])
__global__ __launch_bounds__(256)
void time_enc_kernel(const float* __restrict__ t, const float* __restrict__ w,
                     const float* __restrict__ b, float* __restrict__ out, int n) {
  const int i = blockIdx.x * blockDim.x + threadIdx.x;
  if (i >= n * TDIM) return;
  const int row = i >> 7, col = i & (TDIM - 1);
  out[i] = __cosf(fmaf(t[row], w[col], b[col]));
}

// ---------------------------------------------------------------------------
// Phase C kernel 1: wave per incidence: 384-wide dot -> logit, atomicMax per node
// ---------------------------------------------------------------------------
__global__ __launch_bounds__(256)
void prop_logit_kernel(const float* __restrict__ enew, const float* __restrict__ eteS,
                       const float* __restrict__ nemb, const float* __restrict__ nte,
                       const int* __restrict__ nidx, const int* __restrict__ eseg,
                       const int* __restrict__ emap, float* __restrict__ lws,
                       unsigned* __restrict__ nmax, int M, float scale) {
  const int m = blockIdx.x * 8 + (threadIdx.x >> 5);
  if (m >= M) return;
  const int lane = threadIdx.x & 31;
  const int el = eseg[m];
  const int eg = emap[el];
  const int n = nidx[m];
  const float* ep = enew + (size_t)eg * DDIM;
  const float* np = nemb + (size_t)n * DDIM;
  float p = 0.f;
#pragma unroll
  for (int j = 0; j < 8; j++) { const int d = lane + 32 * j; p = fmaf(ep[d], np[d], p); }
  const float* tp = eteS + (size_t)el * TDIM;
  const float* ntp = nte + (size_t)n * TDIM;
#pragma unroll
  for (int j = 0; j < 4; j++) { const int d = lane + 32 * j; p = fmaf(tp[d], ntp[d], p); }
  p = wred(p) * scale;
  if (lane == 0) {
    lws[m] = p;
    atomicMax(nmax + n, fkey(p));
  }
}

// Phase C kernel 2: thread per incidence: exp(l - max), accumulate denominator
__global__ __launch_bounds__(256)
void prop_expsum_kernel(float* __restrict__ lws, const int* __restrict__ nidx,
                        const unsigned* __restrict__ nmax, float* __restrict__ nsum, int M) {
  const int m = blockIdx.x * blockDim.x + threadIdx.x;
  if (m >= M) return;
  const int n = nidx[m];
  const float ex = __expf(lws[m] - fdec(nmax[n]));
  lws[m] = ex;
  unsafeAtomicAdd(nsum + n, ex);
}

// Phase C kernel 3: wave per incidence: out[n] += sign * attn * enew[eg]
__global__ __launch_bounds__(256)
void prop_accum_kernel(const float* __restrict__ enew, const float* __restrict__ lws,
                       const int* __restrict__ nidx, const int* __restrict__ eseg,
                       const int* __restrict__ emap, const float* __restrict__ nsum,
                       float* __restrict__ outn, int M, float sign) {
  const int m = blockIdx.x * 8 + (threadIdx.x >> 5);
  if (m >= M) return;
  const int lane = threadIdx.x & 31;
  const int n = nidx[m];
  const float wgt = sign * lws[m] / nsum[n];
  const int eg = emap[eseg[m]];
  const float* ep = enew + (size_t)eg * DDIM;
  float* op = outn + (size_t)n * DDIM;
#pragma unroll
  for (int j = 0; j < 8; j++) {
    const int d = lane + 32 * j;
    unsafeAtomicAdd(op + d, wgt * ep[d]);
  }
}

static inline int cdiv(int a, int b) { return (a + b - 1) / b; }

extern "C" void kernel_launch(void* const* d_in, const int* in_sizes, int n_in,
                              void* d_out, int out_size, void* d_ws, size_t ws_size,
                              hipStream_t stream) {
  const float* node_emb      = (const float*)d_in[0];
  const float* edge_emb      = (const float*)d_in[1];
  const float* lin_v_w       = (const float*)d_in[2];
  const float* lin_v_b       = (const float*)d_in[3];
  const float* lin_e_w       = (const float*)d_in[4];
  const float* lin_e_b       = (const float*)d_in[5];
  const float* alpha_w       = (const float*)d_in[6];
  const float* time_w        = (const float*)d_in[7];
  const float* time_b        = (const float*)d_in[8];
  const float* node_time     = (const float*)d_in[9];
  const float* in_edge_time  = (const float*)d_in[10];
  const float* out_edge_time = (const float*)d_in[11];
  const int*   in_emap       = (const int*)d_in[12];   // input_edge_indices
  const int*   out_emap      = (const int*)d_in[13];   // output_edge_indices
  const int*   in_nidx       = (const int*)d_in[14];
  const int*   in_eseg       = (const int*)d_in[15];
  const int*   out_nidx      = (const int*)d_in[16];
  const int*   out_eseg      = (const int*)d_in[17];

  const int N     = in_sizes[0] / DDIM;
  const int E_TOT = in_sizes[1] / DDIM;
  const int E_IN  = in_sizes[12];
  const int E_OUT = in_sizes[13];
  const int M_IN  = in_sizes[14];
  const int M_OUT = in_sizes[16];

  // ---- workspace layout (floats) ----
  float* ws = (float*)d_ws;
  float* nv      = ws;                 ws += (size_t)N * DDIM;
  float* ee      = ws;                 ws += (size_t)E_TOT * DDIM;
  float* enew    = ws;                 ws += (size_t)E_TOT * DDIM;
  float* nte     = ws;                 ws += (size_t)N * TDIM;
  float* ete_in  = ws;                 ws += (size_t)E_IN * TDIM;
  float* ete_out = ws;                 ws += (size_t)E_OUT * TDIM;
  float* lws_in  = ws;                 ws += (size_t)M_IN;
  float* lws_out = ws;                 ws += (size_t)M_OUT;
  unsigned* stats   = (unsigned*)ws;   // [nmax_in | nsum_in | nmax_out | nsum_out]
  unsigned* nmax_in  = stats;
  float*    nsum_in  = (float*)(stats + (size_t)N);
  unsigned* nmax_out = stats + (size_t)2 * N;
  float*    nsum_out = (float*)(stats + (size_t)3 * N);

  float* out_node = (float*)d_out;
  float* out_edge = out_node + (size_t)N * DDIM;

  // ---- init: stats = 0 (valid identity for both key-max and f32 sum) ----
  hipMemsetAsync(stats, 0, sizeof(unsigned) * (size_t)4 * N, stream);
  // node output starts at node_emb; edge output is the ORIGINAL edge_emb
  hipMemcpyAsync(out_node, node_emb, sizeof(float) * (size_t)N * DDIM,
                 hipMemcpyDeviceToDevice, stream);
  hipMemcpyAsync(out_edge, edge_emb, sizeof(float) * (size_t)E_TOT * DDIM,
                 hipMemcpyDeviceToDevice, stream);

  // ---- dense GEMMs on matrix cores ----
  {
    const int waves = (N / 16) * 4;
    gemm256_wmma_kernel<<<cdiv(waves, 8), 256, 0, stream>>>(node_emb, lin_v_w, lin_v_b, nv, N);
  }
  {
    const int waves = (E_TOT / 16) * 4;
    gemm256_wmma_kernel<<<cdiv(waves, 8), 256, 0, stream>>>(edge_emb, lin_e_w, lin_e_b, ee, E_TOT);
  }

  // ---- phase B: per-edge online-softmax aggregation ----
  edge_aggr_kernel<<<cdiv(E_IN, 8), 256, 0, stream>>>(
      edge_emb, ee, nv, node_emb, alpha_w, in_nidx, in_eseg, in_emap, enew, E_IN, M_IN, 0);
  edge_aggr_kernel<<<cdiv(E_OUT, 8), 256, 0, stream>>>(
      edge_emb, ee, nv, node_emb, alpha_w, out_nidx, out_eseg, out_emap, enew, E_OUT, M_OUT, 1);

  // ---- time encodings ----
  time_enc_kernel<<<cdiv(N * TDIM, 256), 256, 0, stream>>>(node_time, time_w, time_b, nte, N);
  time_enc_kernel<<<cdiv(E_IN * TDIM, 256), 256, 0, stream>>>(in_edge_time, time_w, time_b, ete_in, E_IN);
  time_enc_kernel<<<cdiv(E_OUT * TDIM, 256), 256, 0, stream>>>(out_edge_time, time_w, time_b, ete_out, E_OUT);

  // ---- phase C: per-node softmax over incident edges, then scatter ----
  const float scale = 1.0f / sqrtf((float)(DDIM + TDIM));
  prop_logit_kernel<<<cdiv(M_IN, 8), 256, 0, stream>>>(
      enew, ete_in, node_emb, nte, in_nidx, in_eseg, in_emap, lws_in, nmax_in, M_IN, scale);
  prop_logit_kernel<<<cdiv(M_OUT, 8), 256, 0, stream>>>(
      enew, ete_out, node_emb, nte, out_nidx, out_eseg, out_emap, lws_out, nmax_out, M_OUT, scale);

  prop_expsum_kernel<<<cdiv(M_IN, 256), 256, 0, stream>>>(lws_in, in_nidx, nmax_in, nsum_in, M_IN);
  prop_expsum_kernel<<<cdiv(M_OUT, 256), 256, 0, stream>>>(lws_out, out_nidx, nmax_out, nsum_out, M_OUT);

  prop_accum_kernel<<<cdiv(M_IN, 8), 256, 0, stream>>>(
      enew, lws_in, in_nidx, in_eseg, in_emap, nsum_in, out_node, M_IN, +1.0f);
  prop_accum_kernel<<<cdiv(M_OUT, 8), 256, 0, stream>>>(
      enew, lws_out, out_nidx, out_eseg, out_emap, nsum_out, out_node, M_OUT, -1.0f);

  (void)n_in; (void)out_size; (void)ws_size;
}